// ANFIS_26259430048456
// MI455X (gfx1250) — compile-verified
//
#include <hip/hip_runtime.h>
#include <math.h>

typedef __attribute__((ext_vector_type(2))) float v2f;
typedef __attribute__((ext_vector_type(8))) float v8f;

#define N_IN    16
#define N_J     32      // 16 inputs * 2 MFs
#define N_RULES 64
#define KPAD    20      // 17 (x_aug) padded to multiple of 4
#define BATCH   131072

__launch_bounds__(256, 2)
__global__ void anfis_wmma_kernel(const float* __restrict__ x,
                                  const float* __restrict__ center,
                                  const float* __restrict__ log_sigma,
                                  const float* __restrict__ consequent,
                                  const int*   __restrict__ rule_indices,
                                  const float* __restrict__ active_mask,
                                  float* __restrict__ out)
{
    // LDS-staged constants (~13.8 KB of the 320 KB WGP pool)
    __shared__ float sMsel[N_J][N_RULES];   // Msel^T: [j][rule], one-hot select
    __shared__ float sBc[KPAD][N_RULES];    // consequent^T, zero-padded rows 17..19
    __shared__ float sC[N_J];               // centers, flat j = 2*i + m
    __shared__ float sIS[N_J];              // 1 / (exp(log_sigma) + 1e-6)
    __shared__ float sAm[N_RULES];          // active mask

    const int tid = threadIdx.x;

    for (int idx = tid; idx < N_J * N_RULES; idx += blockDim.x) {
        const int j = idx >> 6, r = idx & 63;
        sMsel[j][r] = (rule_indices[r * N_IN + (j >> 1)] == (j & 1)) ? 1.0f : 0.0f;
    }
    for (int idx = tid; idx < KPAD * N_RULES; idx += blockDim.x) {
        const int k = idx >> 6, r = idx & 63;
        sBc[k][r] = (k < 17) ? consequent[r * 17 + k] : 0.0f;
    }
    if (tid < N_J) {
        sC[tid]  = center[tid];
        sIS[tid] = 1.0f / (__expf(log_sigma[tid]) + 1e-6f);
    }
    if (tid < N_RULES) sAm[tid] = active_mask[tid];
    __syncthreads();

    const int lane = tid & 31;
    const int col  = lane & 15;   // rule column within 16-wide tile / batch row within tile
    const int half = lane >> 4;   // lane half selects K-pair in A/B fragments
    const int waveId = (blockIdx.x * blockDim.x + tid) >> 5;
    const int nWaves = (gridDim.x * blockDim.x) >> 5;
    const int nTiles = BATCH / 16;

    // ---- per-lane loop invariants (constant indices after unroll -> no register muxes)
    float c0[8], c1[8], is0[8], is1[8];
    #pragma unroll
    for (int kc = 0; kc < 8; ++kc) {
        const int j0 = 4 * kc + 2 * half;        // runtime LDS address: plain ds_load
        c0[kc]  = sC[j0];   c1[kc]  = sC[j0 + 1];
        is0[kc] = sIS[j0];  is1[kc] = sIS[j0 + 1];
    }
    float am[4];
    float amsum = 0.f;
    #pragma unroll
    for (int nt = 0; nt < 4; ++nt) { am[nt] = sAm[nt * 16 + col]; amsum += am[nt]; }
    #pragma unroll
    for (int m = 1; m < 16; m <<= 1) amsum += __shfl_xor(amsum, m, 32);
    const float amden = fmaxf(amsum, 1.0f);
    float fb[4];                                   // fallback weights (loop-invariant)
    #pragma unroll
    for (int nt = 0; nt < 4; ++nt) fb[nt] = am[nt] / amden;
    const float biasA0 = (half == 0) ? 1.0f : 0.0f;   // x_aug bias element at k=16

    for (int t = waveId; t < nTiles; t += nWaves) {
        const int rowBase = t * 16;
        const float* xrow = x + (size_t)(rowBase + col) * N_IN;
        if (t + nWaves < nTiles)
            __builtin_prefetch(x + (size_t)((t + nWaves) * 16 + col) * N_IN, 0, 0);

        const v8f zero = {0.f,0.f,0.f,0.f,0.f,0.f,0.f,0.f};
        v8f accS[4] = {zero, zero, zero, zero};   // S = q @ Msel^T   (16x32 * 32x64)
        v8f accR[4] = {zero, zero, zero, zero};   // rule_out = x_aug @ Bc (16x20 * 20x64)

        // ---- firing-exponent GEMM: lane's A pair shares one x element (i = 2*kc + half)
        #pragma unroll
        for (int kc = 0; kc < 8; ++kc) {
            const float xv = xrow[2 * kc + half];         // runtime-addressed load, no mux
            const float d0 = (xv - c0[kc]) * is0[kc];
            const float d1 = (xv - c1[kc]) * is1[kc];
            v2f a; a.x = d0 * d0; a.y = d1 * d1;
            const int j0 = 4 * kc + 2 * half;
            #pragma unroll
            for (int nt = 0; nt < 4; ++nt) {
                v2f b; b.x = sMsel[j0][nt * 16 + col]; b.y = sMsel[j0 + 1][nt * 16 + col];
                accS[nt] = __builtin_amdgcn_wmma_f32_16x16x4_f32(
                    false, a, false, b, (short)0, accS[nt], false, false);
            }
        }

        // ---- consequent GEMM, K-steps 0..3: A pair = x[4kc+2h], x[4kc+2h+1] (8B-aligned)
        #pragma unroll
        for (int kc = 0; kc < 4; ++kc) {
            const float2 p = *(const float2*)(xrow + 4 * kc + 2 * half);
            v2f a; a.x = p.x; a.y = p.y;
            const int k0 = 4 * kc + 2 * half;
            #pragma unroll
            for (int nt = 0; nt < 4; ++nt) {
                v2f b; b.x = sBc[k0][nt * 16 + col]; b.y = sBc[k0 + 1][nt * 16 + col];
                accR[nt] = __builtin_amdgcn_wmma_f32_16x16x4_f32(
                    false, a, false, b, (short)0, accR[nt], false, false);
            }
        }
        // ---- K-step 4 (k=16..19): bias=1 at k=16, zeros beyond
        {
            v2f a; a.x = biasA0; a.y = 0.f;
            const int k0 = 16 + 2 * half;
            #pragma unroll
            for (int nt = 0; nt < 4; ++nt) {
                v2f b; b.x = sBc[k0][nt * 16 + col]; b.y = sBc[k0 + 1][nt * 16 + col];
                accR[nt] = __builtin_amdgcn_wmma_f32_16x16x4_f32(
                    false, a, false, b, (short)0, accR[nt], false, false);
            }
        }

        // ---- epilogue: firing = exp(-0.5*S)*am, normalize (with fallback), weighted sum
        // C/D layout: element e of accum <-> batch row (e + 8*half), rule col = nt*16 + col
        #pragma unroll
        for (int e = 0; e < 8; ++e) {
            float f[4]; float ps = 0.f;
            #pragma unroll
            for (int nt = 0; nt < 4; ++nt) {
                f[nt] = __expf(-0.5f * accS[nt][e]) * am[nt];   // single v_exp_f32
                ps += f[nt];
            }
            float rowsum = ps;
            #pragma unroll
            for (int m = 1; m < 16; m <<= 1) rowsum += __shfl_xor(rowsum, m, 32);

            const bool  useFb = (rowsum <= 1e-12f);
            const float inv   = 1.0f / fmaxf(rowsum, 1e-12f);
            float acc = 0.f;
            #pragma unroll
            for (int nt = 0; nt < 4; ++nt) {
                const float wb = useFb ? fb[nt] : (f[nt] * inv);
                acc += wb * accR[nt][e];
            }
            #pragma unroll
            for (int m = 1; m < 16; m <<= 1) acc += __shfl_xor(acc, m, 32);

            if (col == 0) {
                float s = 1.0f / (1.0f + __expf(-acc));
                s = fminf(fmaxf(s, 1e-7f), 1.0f - 1e-7f);
                out[rowBase + e + 8 * half] = s;
            }
        }
    }
}

extern "C" void kernel_launch(void* const* d_in, const int* in_sizes, int n_in,
                              void* d_out, int out_size, void* d_ws, size_t ws_size,
                              hipStream_t stream) {
    const float* x            = (const float*)d_in[0];
    const float* center       = (const float*)d_in[1];
    const float* log_sigma    = (const float*)d_in[2];
    const float* consequent   = (const float*)d_in[3];
    const int*   rule_indices = (const int*)  d_in[4];
    const float* active_mask  = (const float*)d_in[5];
    float* out = (float*)d_out;

    // 512 blocks x 256 threads = 4096 wave32's; 8192 row-tiles -> 2 tiles/wave (uniform)
    anfis_wmma_kernel<<<dim3(512), dim3(256), 0, stream>>>(
        x, center, log_sigma, consequent, rule_indices, active_mask, out);
}